// DGCNN_48550310314213
// MI455X (gfx1250) — compile-verified
//
#include <hip/hip_runtime.h>

typedef float v2f __attribute__((ext_vector_type(2)));
typedef float v8f __attribute__((ext_vector_type(8)));

#define NPTS 4096
#define NB   4
#define KNN  20
#define NEG_INF (-3.402823466e38f)

__device__ __forceinline__ float lrelu(float v) { return v > 0.f ? v : 0.2f * v; }
// order-preserving float <-> uint mapping for atomic max-pool
__device__ __forceinline__ unsigned f2ord(float f) {
    unsigned b = __float_as_uint(f);
    return (b & 0x80000000u) ? ~b : (b | 0x80000000u);
}
__device__ __forceinline__ float ord2f(unsigned u) {
    unsigned b = (u & 0x80000000u) ? (u & 0x7fffffffu) : ~u;
    return __uint_as_float(b);
}

// ---------------------------------------------------------------------------
// Prep: pad/transpose x (B,6,N) -> xt8 (B,N,8) + squared norms (B,N)
// ---------------------------------------------------------------------------
__global__ void prep_euclid(const float* __restrict__ x, float* __restrict__ xt8,
                            float* __restrict__ sqn) {
    int t = blockIdx.x * blockDim.x + threadIdx.x;   // b*N + n
    if (t >= NB * NPTS) return;
    int b = t / NPTS, n = t % NPTS;
    const float* xb = x + (size_t)b * 6 * NPTS + n;
    float s = 0.f;
#pragma unroll
    for (int c = 0; c < 6; ++c) {
        float v = xb[(size_t)c * NPTS];
        s += v * v;
        xt8[(size_t)t * 8 + c] = v;
    }
    xt8[(size_t)t * 8 + 6] = 0.f;
    xt8[(size_t)t * 8 + 7] = 0.f;
    sqn[t] = s;
}

// Normalize 64-channel slice of (B,192,N) into xn (B,N,64)
__global__ void prep_cosine(const float* __restrict__ xin, long inBS,
                            float* __restrict__ xn) {
    int t = blockIdx.x * blockDim.x + threadIdx.x;
    if (t >= NB * NPTS) return;
    int b = t / NPTS, n = t % NPTS;
    const float* xb = xin + (size_t)b * inBS + n;
    float s = 0.f;
    for (int c = 0; c < 64; ++c) {
        float v = xb[(size_t)c * NPTS];
        s += v * v;
    }
    float inv = 1.f / (sqrtf(s) + 1e-8f);
    for (int c = 0; c < 64; ++c)
        xn[(size_t)t * 64 + c] = xb[(size_t)c * NPTS] * inv;
}

// ---------------------------------------------------------------------------
// Fused WMMA Gram + top-k KNN. F: (B,N,Cp) row-major, Cp in {8, 64}.
// Score = F_i . F_j  (- 0.5*sq_j for euclid).  D tile maps M=candidate j,
// N=query i, so each lane owns one query (8 candidates per tile).
// Two interleaved accumulator chains break the WMMA RAW dependency chain.
// grid: (N/64, B), block 128 (4 waves, one 16-query tile per wave)
// ---------------------------------------------------------------------------
__global__ __launch_bounds__(128) void knn_topk(const float* __restrict__ F, int Cp,
                                                const float* __restrict__ sqn,
                                                int* __restrict__ idxOut) {
    __shared__ float Vs[4][16][2 * KNN];
    __shared__ int   Is[4][16][2 * KNN];
    int lane = threadIdx.x & 31;
    int w    = threadIdx.x >> 5;
    int b    = blockIdx.y;
    int i0   = blockIdx.x * 64 + w * 16;
    int l    = lane & 15, half = lane >> 4;
    const float* Fb = F + (size_t)b * NPTS * Cp;
    const float* sq = sqn ? sqn + (size_t)b * NPTS : nullptr;

    float bv[KNN]; int bj[KNN];
#pragma unroll
    for (int t = 0; t < KNN; ++t) { bv[t] = NEG_INF; bj[t] = 0x7fffffff; }
    float wv = NEG_INF; int wslot = 0;

    const float* Bi = Fb + (size_t)(i0 + l) * Cp;     // this lane's query row
    for (int j0 = 0; j0 < NPTS; j0 += 16) {
        const float* Aj = Fb + (size_t)(j0 + l) * Cp; // candidate rows
        v8f acc0 = {}, acc1 = {};
        for (int k0 = 0; k0 < Cp; k0 += 8) {          // 2 chains of K=4 steps
            int ka = k0 + 2 * half;
            v2f a0, b0, a1, b1;
            a0.x = Aj[ka];     a0.y = Aj[ka + 1];
            b0.x = Bi[ka];     b0.y = Bi[ka + 1];
            a1.x = Aj[ka + 4]; a1.y = Aj[ka + 5];
            b1.x = Bi[ka + 4]; b1.y = Bi[ka + 5];
            acc0 = __builtin_amdgcn_wmma_f32_16x16x4_f32(
                false, a0, false, b0, (short)0, acc0, false, false);
            acc1 = __builtin_amdgcn_wmma_f32_16x16x4_f32(
                false, a1, false, b1, (short)0, acc1, false, false);
        }
        v8f acc = acc0 + acc1;
#pragma unroll
        for (int r = 0; r < 8; ++r) {
            int   j = j0 + r + 8 * half;
            float v = acc[r];
            if (sq) v -= 0.5f * sq[j];
            if (v > wv) {               // ascending j => ties keep lower index
                bv[wslot] = v; bj[wslot] = j;
                wv = bv[0]; wslot = 0;
#pragma unroll
                for (int t = 1; t < KNN; ++t)
                    if (bv[t] < wv) { wv = bv[t]; wslot = t; }
            }
        }
    }
#pragma unroll
    for (int t = 0; t < KNN; ++t) {
        Vs[w][l][half * KNN + t] = bv[t];
        Is[w][l][half * KNN + t] = bj[t];
    }
    __syncthreads();
    if (lane < 16) {      // merge the two half-lists, emit sorted top-20
        int q = i0 + lane;
        float* V = Vs[w][lane];
        int*   I = Is[w][lane];
        int* orow = idxOut + ((size_t)b * NPTS + q) * KNN;
        for (int t = 0; t < KNN; ++t) {
            float best = NEG_INF; int bidx = 0x7fffffff, bs = 0;
            for (int s = 0; s < 2 * KNN; ++s) {
                float v = V[s]; int jj = I[s];
                if (v > best || (v == best && jj < bidx)) { best = v; bidx = jj; bs = s; }
            }
            V[bs] = NEG_INF; I[bs] = 0x7fffffff;
            orow[t] = bidx;
        }
    }
}

// ---------------------------------------------------------------------------
// EdgeConv: per point, gather k neighbors, 2 fused 1x1 convs + lrelu, max-k.
// grid (N, B), block 64 (thread = output channel)
// ---------------------------------------------------------------------------
__global__ __launch_bounds__(64) void edge_conv(
    const float* __restrict__ Xin, long inBS, int C, const int* __restrict__ idx,
    const float* __restrict__ w0, const float* __restrict__ s0, const float* __restrict__ o0,
    const float* __restrict__ w1, const float* __restrict__ s1, const float* __restrict__ o1,
    float* __restrict__ Yout, long outBS) {
    __shared__ float w0s[64 * 128];
    __shared__ float w1s[64 * 64];
    __shared__ float xi[64];
    __shared__ float xj[KNN * 64];
    __shared__ float h0s[KNN * 64];
    __shared__ int   nbr[KNN];
    int tid = threadIdx.x;
    int n = blockIdx.x, b = blockIdx.y;
    int C2 = 2 * C;
    for (int e = tid; e < 64 * C2; e += 64) w0s[e] = w0[e];
    for (int e = tid; e < 64 * 64; e += 64) w1s[e] = w1[e];
    const float* xb = Xin + (size_t)b * inBS;
    if (tid < C)   xi[tid]  = xb[(size_t)tid * NPTS + n];
    if (tid < KNN) nbr[tid] = idx[((size_t)b * NPTS + n) * KNN + tid];
    __syncthreads();
    for (int e = tid; e < KNN * C; e += 64) {
        int kk = e / C, c = e - kk * C;
        xj[e] = xb[(size_t)c * NPTS + nbr[kk]];
    }
    __syncthreads();
    {   // layer 0: (64 x 2C) on edge feats [xj-xi, xi]
        int o = tid;
        float sc = s0[o], of = o0[o];
        for (int kk = 0; kk < KNN; ++kk) {
            float acc = 0.f;
            for (int c = 0; c < C; ++c) {
                float cv = xi[c];
                acc += w0s[o * C2 + c] * (xj[kk * C + c] - cv);
                acc += w0s[o * C2 + C + c] * cv;
            }
            h0s[kk * 64 + o] = lrelu(acc * sc + of);
        }
    }
    __syncthreads();
    {   // layer 1: (64 x 64) + max over k
        int o = tid;
        float sc = s1[o], of = o1[o];
        float best = NEG_INF;
        for (int kk = 0; kk < KNN; ++kk) {
            float acc = 0.f;
            for (int c = 0; c < 64; ++c) acc += w1s[o * 64 + c] * h0s[kk * 64 + c];
            best = fmaxf(best, lrelu(acc * sc + of));
        }
        Yout[(size_t)b * outBS + (size_t)o * NPTS + n] = best;
    }
}

// ---------------------------------------------------------------------------
// Generic f32 WMMA GEMM:  Y[b] = act( (W . X[b]) * scale + offset )
// W: M x K (row stride ldw), X: K x NPTS per batch.
// One wave computes a 16(M) x 64(N) tile: the A (weight) fragment is loaded
// once per K-step and reused across 4 WMMAs (4 independent accumulators).
// doPool: skip store, fold global max over n into poolU via ordered atomicMax.
// ---------------------------------------------------------------------------
__global__ __launch_bounds__(32) void wmma_gemm(
    const float* __restrict__ W, const float* __restrict__ X, long xBS,
    int M, int K, int ldw,
    const float* __restrict__ scale, const float* __restrict__ offset, int offPerBatch,
    int doAct, float* __restrict__ Y, long yBS, unsigned* __restrict__ poolU) {
    int lane = threadIdx.x & 31;
    int l = lane & 15, half = lane >> 4;
    int nb = blockIdx.x * 64, mb = blockIdx.y * 16, b = blockIdx.z;
    const float* Xb = X + (size_t)b * xBS;
    int arow = mb + l;
    bool avalid = arow < M;
    const float* Wr = W + (size_t)(avalid ? arow : 0) * ldw;
    v8f acc[4] = {{}, {}, {}, {}};
    for (int k0 = 0; k0 < K; k0 += 4) {
        int ka = k0 + 2 * half;
        v2f a;
        a.x = avalid ? Wr[ka] : 0.f;
        a.y = avalid ? Wr[ka + 1] : 0.f;
        const float* x0 = Xb + (size_t)ka * NPTS + nb + l;
        const float* x1 = Xb + (size_t)(ka + 1) * NPTS + nb + l;
#pragma unroll
        for (int u = 0; u < 4; ++u) {
            v2f bb;
            bb.x = x0[16 * u];
            bb.y = x1[16 * u];
            acc[u] = __builtin_amdgcn_wmma_f32_16x16x4_f32(
                false, a, false, bb, (short)0, acc[u], false, false);
        }
    }
#pragma unroll
    for (int u = 0; u < 4; ++u) {
        int n = nb + 16 * u + l;
#pragma unroll
        for (int r = 0; r < 8; ++r) {
            int m = mb + r + 8 * half;
            int mc = m < M ? m : 0;
            float v = acc[u][r];
            if (scale)  v *= scale[mc];
            if (offset) v += offset[(offPerBatch ? (size_t)b * M : 0) + mc];
            if (doAct)  v = lrelu(v);
            if (poolU) {
                float mv = v;   // max over the 16 n's held by this half-wave
                for (int d = 1; d < 16; d <<= 1) mv = fmaxf(mv, __shfl_xor(mv, d, 32));
                if (l == 0 && m < M) atomicMax(&poolU[(size_t)b * M + m], f2ord(mv));
            } else if (m < M) {
                Y[(size_t)b * yBS + (size_t)m * NPTS + n] = v;
            }
        }
    }
}

// ---------------------------------------------------------------------------
__global__ void zero_u32(unsigned* p, int n) {
    int t = blockIdx.x * blockDim.x + threadIdx.x;
    if (t < n) p[t] = 0u;
}
__global__ void decode_g(const unsigned* __restrict__ gu, float* __restrict__ g, int n) {
    int t = blockIdx.x * blockDim.x + threadIdx.x;
    if (t < n) g[t] = ord2f(gu[t]);
}
// offEff[b,o] = sf1[o] * (wf1[o,192:1216] . g[b]) + of1[o]   (g is broadcast over n)
__global__ void gproj(const float* __restrict__ wf1, const float* __restrict__ g,
                      const float* __restrict__ sf1, const float* __restrict__ of1,
                      float* __restrict__ offEff) {
    int t = blockIdx.x * blockDim.x + threadIdx.x;   // b*512 + o
    if (t >= NB * 512) return;
    int b = t / 512, o = t % 512;
    const float* wr = wf1 + (size_t)o * 1216 + 192;
    const float* gb = g + (size_t)b * 1024;
    float acc = 0.f;
    for (int c = 0; c < 1024; ++c) acc += wr[c] * gb[c];
    offEff[t] = acc * sf1[o] + of1[o];
}

// ---------------------------------------------------------------------------
extern "C" void kernel_launch(void* const* d_in, const int* in_sizes, int n_in,
                              void* d_out, int out_size, void* d_ws, size_t ws_size,
                              hipStream_t stream) {
    const float* x    = (const float*)d_in[0];
    const float* w1_0 = (const float*)d_in[1];
    const float* s1_0 = (const float*)d_in[2];
    const float* o1_0 = (const float*)d_in[3];
    const float* w1_1 = (const float*)d_in[4];
    const float* s1_1 = (const float*)d_in[5];
    const float* o1_1 = (const float*)d_in[6];
    const float* w2_0 = (const float*)d_in[7];
    const float* s2_0 = (const float*)d_in[8];
    const float* o2_0 = (const float*)d_in[9];
    const float* w2_1 = (const float*)d_in[10];
    const float* s2_1 = (const float*)d_in[11];
    const float* o2_1 = (const float*)d_in[12];
    const float* w3_0 = (const float*)d_in[13];
    const float* s3_0 = (const float*)d_in[14];
    const float* o3_0 = (const float*)d_in[15];
    const float* w3_1 = (const float*)d_in[16];
    const float* s3_1 = (const float*)d_in[17];
    const float* o3_1 = (const float*)d_in[18];
    const float* w4   = (const float*)d_in[19];
    const float* b4   = (const float*)d_in[20];
    const float* wf1  = (const float*)d_in[21];
    const float* sf1  = (const float*)d_in[22];
    const float* of1  = (const float*)d_in[23];
    const float* wf2  = (const float*)d_in[24];
    const float* sf2  = (const float*)d_in[25];
    const float* of2  = (const float*)d_in[26];
    const float* wf3  = (const float*)d_in[27];
    // d_in[28] = k (== 20, compiled in)

    char*  ws  = (char*)d_ws;
    size_t off = 0;
    auto alloc = [&](size_t bytes) -> void* {
        void* p = ws + off;
        off += (bytes + 255) & ~(size_t)255;
        return p;
    };
    float*    xt8  = (float*)alloc((size_t)NB * NPTS * 8 * 4);
    float*    sqn  = (float*)alloc((size_t)NB * NPTS * 4);
    float*    xn   = (float*)alloc((size_t)NB * NPTS * 64 * 4);
    int*      idx  = (int*)alloc((size_t)NB * NPTS * KNN * 4);
    float*    xc   = (float*)alloc((size_t)NB * 192 * NPTS * 4);   // x1|x2|x3
    unsigned* gu   = (unsigned*)alloc((size_t)NB * 1024 * 4);
    float*    g    = (float*)alloc((size_t)NB * 1024 * 4);
    float*    offE = (float*)alloc((size_t)NB * 512 * 4);
    float*    h1   = (float*)alloc((size_t)NB * 512 * NPTS * 4);
    float*    h2   = (float*)alloc((size_t)NB * 256 * NPTS * 4);

    const long xcBS = (long)192 * NPTS;
    int  tot = NB * NPTS;
    dim3 kg(NPTS / 64, NB);
    dim3 eg(NPTS, NB);

    // stage 1: euclid knn on raw x, edge conv 1 -> xc[:,0:64]
    prep_euclid<<<(tot + 127) / 128, 128, 0, stream>>>(x, xt8, sqn);
    knn_topk<<<kg, 128, 0, stream>>>(xt8, 8, sqn, idx);
    edge_conv<<<eg, 64, 0, stream>>>(x, (long)6 * NPTS, 6, idx,
                                     w1_0, s1_0, o1_0, w1_1, s1_1, o1_1, xc, xcBS);
    // stage 2: cosine knn on x1, edge conv 2 -> xc[:,64:128]
    prep_cosine<<<(tot + 127) / 128, 128, 0, stream>>>(xc, xcBS, xn);
    knn_topk<<<kg, 128, 0, stream>>>(xn, 64, nullptr, idx);
    edge_conv<<<eg, 64, 0, stream>>>(xc, xcBS, 64, idx,
                                     w2_0, s2_0, o2_0, w2_1, s2_1, o2_1,
                                     xc + (size_t)64 * NPTS, xcBS);
    // stage 3: cosine knn on x2, edge conv 3 -> xc[:,128:192]
    prep_cosine<<<(tot + 127) / 128, 128, 0, stream>>>(xc + (size_t)64 * NPTS, xcBS, xn);
    knn_topk<<<kg, 128, 0, stream>>>(xn, 64, nullptr, idx);
    edge_conv<<<eg, 64, 0, stream>>>(xc + (size_t)64 * NPTS, xcBS, 64, idx,
                                     w3_0, s3_0, o3_0, w3_1, s3_1, o3_1,
                                     xc + (size_t)128 * NPTS, xcBS);

    // x4 = lrelu(w4.xc + b4), fused global max pool over n -> g
    zero_u32<<<(NB * 1024 + 255) / 256, 256, 0, stream>>>(gu, NB * 1024);
    wmma_gemm<<<dim3(NPTS / 64, 64, NB), 32, 0, stream>>>(
        w4, xc, xcBS, 1024, 192, 192, nullptr, b4, 0, 1, nullptr, 0, gu);
    decode_g<<<(NB * 1024 + 255) / 256, 256, 0, stream>>>(gu, g, NB * 1024);

    // fold wf1's g-columns into a per-batch effective offset
    gproj<<<(NB * 512 + 127) / 128, 128, 0, stream>>>(wf1, g, sf1, of1, offE);

    // h1 = lrelu((wf1[:, :192].xc)*sf1 + offE)    (wf1 row stride 1216)
    wmma_gemm<<<dim3(NPTS / 64, 32, NB), 32, 0, stream>>>(
        wf1, xc, xcBS, 512, 192, 1216, sf1, offE, 1, 1, h1, (long)512 * NPTS, nullptr);
    // h2 = lrelu((wf2.h1)*sf2 + of2)
    wmma_gemm<<<dim3(NPTS / 64, 16, NB), 32, 0, stream>>>(
        wf2, h1, (long)512 * NPTS, 256, 512, 512, sf2, of2, 0, 1, h2, (long)256 * NPTS, nullptr);
    // out = wf3.h2   (M=13, tile-padded)
    wmma_gemm<<<dim3(NPTS / 64, 1, NB), 32, 0, stream>>>(
        wf3, h2, (long)256 * NPTS, 13, 256, 256, nullptr, nullptr, 0, 0,
        (float*)d_out, (long)13 * NPTS, nullptr);
}